// ContextKnowledgeEncoder_20847771255424
// MI455X (gfx1250) — compile-verified
//
#include <hip/hip_runtime.h>
#include <hip/hip_bf16.h>
#include <math.h>

// ---------------------------------------------------------------------------
// ContextKnowledgeEncoder for MI455X (gfx1250, wave32)
// Memory-bound gather/pool pipeline; ck_attn batched-mv done with exact-f32
// V_WMMA_F32_16X16X4_F32 (A = know_use 16x4 chunk, B = ctx_use chunk
// broadcast across all 16 columns, accumulate over D in steps of 4).
// ---------------------------------------------------------------------------

typedef __attribute__((ext_vector_type(2))) float v2f;
typedef __attribute__((ext_vector_type(8))) float v8f;

#define N_  64
#define TC_ 256
#define K_  32
#define TK_ 64
#define D_  256
#define TFULL_ (TK_ + TC_)

// ---- Kernel 1: ctx_use[n,d] = sum_t mask * embed[src[n,t], d] / sqrt(D*len)
__global__ __launch_bounds__(256) void ctx_pool_kernel(
    const int* __restrict__ src_tokens, const float* __restrict__ embed,
    float* __restrict__ ctx_use) {
  const int n = blockIdx.x;
  const int d = threadIdx.x;            // 0..255
  const int* toks = src_tokens + n * TC_;
  float sum = 0.f;
  int   len = 0;
#pragma unroll 4
  for (int t = 0; t < TC_; ++t) {
    int tok = toks[t];
    if (tok != 0) {
      sum += embed[(size_t)tok * D_ + d];
      ++len;
    }
  }
  float denom = sqrtf((float)D_ * (float)(len > 0 ? len : 1));
  ctx_use[n * D_ + d] = sum / denom;
}

// ---- Kernel 2: know_use[n,k,d] with mask = (tok!=0) & ck_mask[n,k]
__global__ __launch_bounds__(256) void know_pool_kernel(
    const int* __restrict__ know_tokens, const int* __restrict__ ck_mask,
    const float* __restrict__ embed, float* __restrict__ know_use) {
  const int b = blockIdx.x;             // n*K + k
  const int d = threadIdx.x;
  const int cm = ck_mask[b];
  const int* toks = know_tokens + (size_t)b * TK_;
  float sum = 0.f;
  int   len = 0;
#pragma unroll 4
  for (int t = 0; t < TK_; ++t) {
    int tok = toks[t];
    int m = (tok != 0) & (cm != 0);
    if (m) {
      sum += embed[(size_t)tok * D_ + d];
      ++len;
    }
  }
  float denom = sqrtf((float)D_ * (float)(len > 0 ? len : 1));
  know_use[(size_t)b * D_ + d] = sum / denom;
}

// ---- Kernel 3: scores[n,k] = dot(know_use[n,k,:], ctx_use[n,:]) via WMMA f32
// One wave per (n, ktile). A: 16 rows of know_use, 4 K-elems per step.
// B: ctx_use chunk broadcast to all 16 columns -> every column of D equals
// the desired score, so only the K-dim lane striping matters.
__global__ __launch_bounds__(32) void ck_attn_wmma_kernel(
    const float* __restrict__ know_use, const float* __restrict__ ctx_use,
    float* __restrict__ scores) {
  const int blk  = blockIdx.x;          // n*2 + tile
  const int n    = blk >> 1;
  const int tile = blk & 1;
  const int lane = threadIdx.x;         // 0..31 (wave32)
  const int row  = lane & 15;           // M index
  const int koff = (lane >> 4) * 2;     // VGPR K striping: lanes16-31 hold K=2,3

  const float* arow = know_use + ((size_t)n * K_ + tile * 16 + row) * D_;
  const float* bvec = ctx_use + (size_t)n * D_;

  v8f c = {};
  for (int d0 = 0; d0 < D_; d0 += 4) {
    v2f a = *(const v2f*)(arow + d0 + koff);   // A[row, d0+koff .. +1]
    v2f b = *(const v2f*)(bvec + d0 + koff);   // B[d0+koff .., all cols] bcast
    c = __builtin_amdgcn_wmma_f32_16x16x4_f32(
        /*neg_a=*/false, a, /*neg_b=*/false, b,
        /*c_mod=*/(short)0, c, /*reuse_a=*/false, /*reuse_b=*/false);
  }
  // C/D layout: VGPR r, lanes 0-15 -> M=r, col=lane; lanes 16-31 -> M=8+r.
  // Take column 0 (lane 0) and column 0 of the upper half (lane 16).
  if (lane == 0) {
#pragma unroll
    for (int r = 0; r < 8; ++r) scores[n * K_ + tile * 16 + r] = c[r];
  } else if (lane == 16) {
#pragma unroll
    for (int r = 0; r < 8; ++r) scores[n * K_ + tile * 16 + 8 + r] = c[r];
  }
}

// ---- Kernel 4: per-n wave32 argmax (first-max tie-break), cs selection,
// masked ck_attn output.
__global__ __launch_bounds__(32) void select_kernel(
    const float* __restrict__ scores, const int* __restrict__ ck_mask,
    const int* __restrict__ cs_ids, const int* __restrict__ use_cs_ids,
    float* __restrict__ ck_attn_out, int* __restrict__ cs_out) {
  const int n = blockIdx.x;
  const int k = threadIdx.x;            // 0..31 == K
  const float s = scores[n * K_ + k];
  const int   m = ck_mask[n * K_ + k];

  ck_attn_out[n * K_ + k] = m ? s : 0.0f;

  float v = m ? s : -INFINITY;
  int best = k;
#pragma unroll
  for (int off = 16; off > 0; off >>= 1) {
    float ov = __shfl_down(v, off, 32);
    int   ob = __shfl_down(best, off, 32);
    if (ov > v || (ov == v && ob < best)) { v = ov; best = ob; }
  }
  if (k == 0) {
    cs_out[n] = (use_cs_ids[0] != 0) ? cs_ids[n] : best;
  }
}

// ---- Kernel 5: full_enc[n, t<TK, :] = masked gather of selected know span
__global__ __launch_bounds__(256) void gather_know_kernel(
    const int* __restrict__ know_tokens, const int* __restrict__ ck_mask,
    const int* __restrict__ cs, const float* __restrict__ embed,
    float* __restrict__ full_enc, float* __restrict__ full_mask) {
  const int b = blockIdx.x;             // n*TK + t
  const int n = b / TK_;
  const int t = b % TK_;
  const int d = threadIdx.x;
  const int c = cs[n];
  const int tok = know_tokens[((size_t)n * K_ + c) * TK_ + t];
  const int m = (tok != 0) & (ck_mask[n * K_ + c] != 0);
  float val = m ? embed[(size_t)tok * D_ + d] : 0.0f;
  full_enc[((size_t)n * TFULL_ + t) * D_ + d] = val;
  if (d == 0) full_mask[n * TFULL_ + t] = m ? 1.0f : 0.0f;
}

// ---- Kernel 6: full_enc[n, TK+t, :] = masked context gather
__global__ __launch_bounds__(256) void gather_ctx_kernel(
    const int* __restrict__ src_tokens, const float* __restrict__ embed,
    float* __restrict__ full_enc, float* __restrict__ full_mask) {
  const int b = blockIdx.x;             // n*TC + t
  const int n = b / TC_;
  const int t = b % TC_;
  const int d = threadIdx.x;
  const int tok = src_tokens[n * TC_ + t];
  const int m = (tok != 0);
  float val = m ? embed[(size_t)tok * D_ + d] : 0.0f;
  full_enc[((size_t)n * TFULL_ + TK_ + t) * D_ + d] = val;
  if (d == 0) full_mask[n * TFULL_ + TK_ + t] = m ? 1.0f : 0.0f;
}

extern "C" void kernel_launch(void* const* d_in, const int* in_sizes, int n_in,
                              void* d_out, int out_size, void* d_ws, size_t ws_size,
                              hipStream_t stream) {
  (void)in_sizes; (void)n_in; (void)out_size; (void)ws_size;

  const int*   src_tokens  = (const int*)d_in[0];   // [N,TC]
  const int*   know_tokens = (const int*)d_in[1];   // [N,K,TK]
  const int*   ck_mask     = (const int*)d_in[2];   // [N,K]
  const int*   cs_ids      = (const int*)d_in[3];   // [N]
  const int*   use_cs_ids  = (const int*)d_in[4];   // [1]
  const float* embed       = (const float*)d_in[5]; // [V,D]

  // d_out: full_enc [N,TK+TC,D] || full_mask [N,TK+TC] || ck_attn [N,K]
  float* full_enc  = (float*)d_out;
  float* full_mask = full_enc + (size_t)N_ * TFULL_ * D_;
  float* ck_attn   = full_mask + (size_t)N_ * TFULL_;

  // Workspace layout
  char*  ws       = (char*)d_ws;
  float* ctx_use  = (float*)(ws);                              // N*D
  float* know_use = (float*)(ws + 64 * 1024);                  // N*K*D
  float* scores   = (float*)(ws + 64 * 1024 + 2 * 1024 * 1024); // N*K
  int*   cs       = (int*)  (ws + 64 * 1024 + 2 * 1024 * 1024 + 16 * 1024);

  ctx_pool_kernel<<<N_, 256, 0, stream>>>(src_tokens, embed, ctx_use);
  know_pool_kernel<<<N_ * K_, 256, 0, stream>>>(know_tokens, ck_mask, embed, know_use);
  ck_attn_wmma_kernel<<<N_ * 2, 32, 0, stream>>>(know_use, ctx_use, scores);
  select_kernel<<<N_, 32, 0, stream>>>(scores, ck_mask, cs_ids, use_cs_ids, ck_attn, cs);
  gather_know_kernel<<<N_ * TK_, 256, 0, stream>>>(know_tokens, ck_mask, cs, embed,
                                                   full_enc, full_mask);
  gather_ctx_kernel<<<N_ * TC_, 256, 0, stream>>>(src_tokens, embed, full_enc, full_mask);
}